// SplitNetwork_9182640079048
// MI455X (gfx1250) — compile-verified
//
#include <hip/hip_runtime.h>
#include <math.h>

// ---------------- problem constants ----------------
#define N_ATOMS 768
#define HID     112            // hidden dim 100 padded to 7*16
#define HID_P   114            // activation row stride (bank-conflict-free A frags)
#define K2      (HID / 2)      // 56 float2 K-pairs
#define ROWS    128            // pairs per block
#define BLOCK_THREADS 256      // 8 wave32
// hA + hB (ROWS*HID_P each) + Wp (K2*HID float2) + bias + geo
#define SMEM_FLOATS (2*ROWS*HID_P + K2*HID*2 + HID + ROWS*8)

typedef float v2f __attribute__((ext_vector_type(2)));
typedef float v8f __attribute__((ext_vector_type(8)));

__device__ constexpr int IOFF[7] = {0,1,4,9,16,25,36};
// sqrt((2l+1)(l-|m|)!/(l+|m|)!)  (x sqrt(2) for m != 0), in comps order l, m=-l..l
__device__ constexpr float SH_NORM[36] = {
  1.0f,
  1.7320508f, 1.7320508f, 1.7320508f,
  0.6454972f, 1.2909944f, 2.2360680f, 1.2909944f, 0.6454972f,
  0.1394333f, 0.3415650f, 1.0801234f, 2.6457513f, 1.0801234f, 0.3415650f, 0.1394333f,
  0.0211289f, 0.0597614f, 0.2236068f, 0.9486833f, 3.0000000f, 0.9486833f, 0.2236068f, 0.0597614f, 0.0211289f,
  0.0024623f, 0.0077863f, 0.0330343f, 0.1618347f, 0.8563488f, 3.3166248f, 0.8563488f, 0.1618347f, 0.0330343f, 0.0077863f, 0.0024623f
};
__device__ constexpr float RSQL[6] = {1.0f, 0.57735026919f, 0.44721359550f,
                                      0.37796447301f, 0.33333333333f, 0.30151134458f};

__device__ __forceinline__ float ssp(float x) {
  float z = 5.0f * x;
  float sp = fmaxf(z, 0.0f) + log1pf(expf(-fabsf(z)));
  return (sp - 0.69314718056f) * 0.2f;
}

// ---------------- real spherical harmonics (LMAX=5), fully unrolled -> registers
__device__ __forceinline__ void compute_sph(float dx, float dy, float dz, bool mask, float* Y) {
  float xs = mask ? dx : 1.0f;
  float ys = mask ? dy : 0.0f;
  float z  = dz;
  float st = sqrtf(fmaxf(1.0f - z * z, 0.0f));
  float hyp = sqrtf(xs * xs + ys * ys);
  float cphi, sphi;
  if (hyp > 1e-20f) { float ih = 1.0f / hyp; cphi = xs * ih; sphi = ys * ih; }
  else              { cphi = 1.0f; sphi = 0.0f; }

  float P[6][6];
  P[0][0] = 1.0f;
#pragma unroll
  for (int m = 1; m < 6; ++m) P[m][m] = -(float)(2*m-1) * st * P[m-1][m-1];
#pragma unroll
  for (int m = 0; m < 5; ++m) P[m+1][m] = (float)(2*m+1) * z * P[m][m];
#pragma unroll
  for (int m = 0; m < 6; ++m) {
#pragma unroll
    for (int l = m + 2; l < 6; ++l)
      P[l][m] = ((float)(2*l-1) * z * P[l-1][m] - (float)(l+m-1) * P[l-2][m]) * (1.0f/(float)(l-m));
  }
  float cm[6], sn[6];
  cm[0] = 1.0f; sn[0] = 0.0f; cm[1] = cphi; sn[1] = sphi;
#pragma unroll
  for (int m = 2; m < 6; ++m) {
    cm[m] = 2.0f * cphi * cm[m-1] - cm[m-2];
    sn[m] = 2.0f * cphi * sn[m-1] - sn[m-2];
  }
  int idx = 0;
#pragma unroll
  for (int l = 0; l < 6; ++l) {
#pragma unroll
    for (int mm = -l; mm <= l; ++mm) {
      int am = mm < 0 ? -mm : mm;
      float c;
      if      (mm < 0)  c = SH_NORM[idx] * sn[am] * P[l][am];
      else if (mm == 0) c = SH_NORM[idx] * P[l][0];
      else              c = SH_NORM[idx] * cm[am] * P[l][am];
      if (l > 0 && !mask) c = 0.0f;
      Y[idx++] = c;
    }
  }
}

// ---------------- stage weights into LDS in WMMA-fragment order:
// Wp[k2*HID + n] = float2{ W[2*k2][n], W[2*k2+1][n] }  (zero-padded)
__device__ __forceinline__ void load_weights(v2f* __restrict__ Wp, float* __restrict__ bsh,
                                             const float* __restrict__ W,
                                             const float* __restrict__ b,
                                             int din, int dout, int tid) {
  for (int e = tid; e < K2 * HID; e += BLOCK_THREADS) {
    int k2 = e / HID, n = e % HID;
    int k = 2 * k2;
    v2f w;
    w.x = (k     < din && n < dout) ? W[k * dout + n]       : 0.0f;
    w.y = (k + 1 < din && n < dout) ? W[(k + 1) * dout + n] : 0.0f;
    Wp[e] = w;
  }
  for (int n = tid; n < HID; n += BLOCK_THREADS) bsh[n] = (n < dout) ? b[n] : 0.0f;
}

// ---------------- one dense layer: dst[ROWS][HID] = act(src[ROWS][HID] @ W + b)
// Each wave32 owns a 16-row slab, NT tiles of 16 cols, K=HID via V_WMMA_F32_16X16X4_F32.
// A frag: lanes 0-15 rows, K {0,1}; lanes 16-31 K {2,3}  -> one ds_load_b64
// B frag: lane n = r, same K split                        -> one ds_load_b64 (Wp layout)
template <int NT, bool ACT>
__device__ __forceinline__ void gemm_layer(const float* __restrict__ src,
                                           float* __restrict__ dst,
                                           const v2f* __restrict__ Wp,
                                           const float* __restrict__ bsh,
                                           int wave, int lane) {
  const int rowBase = wave * 16;
  const int r   = lane & 15;          // row (A) / col (B,C,D) within tile
  const int kh2 = lane >> 4;          // K-pair sub-offset: 0 or 1
  v8f acc[NT];
#pragma unroll
  for (int nt = 0; nt < NT; ++nt) {
#pragma unroll
    for (int v = 0; v < 8; ++v) acc[nt][v] = 0.0f;
  }
  const float* aRow = src + (rowBase + r) * HID_P;
  for (int k2 = 0; k2 < K2; k2 += 2) {         // K advances by 4 per iter
    const int kk = k2 + kh2;
    v2f a = *(const v2f*)(aRow + kk * 2);      // ds_load_b64
    const v2f* wrow = Wp + kk * HID + r;
#pragma unroll
    for (int nt = 0; nt < NT; ++nt) {
      v2f b = wrow[nt * 16];                   // ds_load_b64
      acc[nt] = __builtin_amdgcn_wmma_f32_16x16x4_f32(
          false, a, false, b, (short)0, acc[nt], false, false);
    }
  }
  const int mAdd = (lane >> 4) * 8;            // C/D: lanes16-31 hold rows M+8
#pragma unroll
  for (int nt = 0; nt < NT; ++nt) {
#pragma unroll
    for (int v = 0; v < 8; ++v) {
      const int n = nt * 16 + r;
      const int m = rowBase + v + mAdd;
      float val = acc[nt][v] + bsh[n];
      dst[m * HID_P + n] = ACT ? ssp(val) : val;
    }
  }
}

// ---------------- radial-MLP front end shared by both conv kernels
// Fills hA with layer1 output, runs layers 2..4, leaving R in hB[ROWS][HID_P].
template <int NT4>
__device__ __forceinline__ void run_radial_mlp(float* hA, float* hB, v2f* Wp, float* bsh,
                                               const float* geo,
                                               const float* W1, const float* b1,
                                               const float* W2, const float* b2,
                                               const float* W3, const float* b3,
                                               const float* W4, const float* b4,
                                               int dout4, int tid, int wave, int lane) {
  // layer 1 (3 -> 100) on VALU: trivial FLOPs
  for (int e = tid; e < ROWS * HID; e += BLOCK_THREADS) {
    int p = e / HID, j = e % HID;
    float v = 0.0f;
    if (j < 100) {
      float g0 = geo[p * 8 + 4], g1 = geo[p * 8 + 5], g2 = geo[p * 8 + 6];
      v = ssp(b1[j] + g0 * W1[j] + g1 * W1[100 + j] + g2 * W1[200 + j]);
    }
    hA[p * HID_P + j] = v;
  }
  load_weights(Wp, bsh, W2, b2, 100, 100, tid);
  __syncthreads();
  gemm_layer<7, true>(hA, hB, Wp, bsh, wave, lane);
  __syncthreads();
  load_weights(Wp, bsh, W3, b3, 100, 100, tid);
  __syncthreads();
  gemm_layer<7, true>(hB, hA, Wp, bsh, wave, lane);
  __syncthreads();
  load_weights(Wp, bsh, W4, b4, 100, dout4, tid);
  __syncthreads();
  gemm_layer<NT4, false>(hA, hB, Wp, bsh, wave, lane);  // R -> hB
  __syncthreads();
}

// ================= kernel 1: fused conv_scalar_in over 128 pairs of row a =================
__global__ void __launch_bounds__(BLOCK_THREADS)
rad1_conv_in_kernel(const float* __restrict__ x,    // [768,4]
                    const float* __restrict__ pos,  // [768,3]
                    const float* W1, const float* b1, const float* W2, const float* b2,
                    const float* W3, const float* b3, const float* W4, const float* b4,
                    float* __restrict__ h_pre) {    // [768,44] (41 used), accumulated
  extern __shared__ float sm[];
  float* hA  = sm;                       // ROWS*HID_P
  float* hB  = hA + ROWS * HID_P;        // ROWS*HID_P
  v2f*   Wp  = (v2f*)(hB + ROWS * HID_P);// K2*HID float2
  float* bsh = (float*)(Wp + K2 * HID);  // HID
  float* geo = bsh + HID;                // ROWS*8: dirs(3), rlen, basis(3), pad

  const int tid = threadIdx.x, wave = tid >> 5, lane = tid & 31;
  const int a = blockIdx.y;
  const int colBase = blockIdx.x * ROWS;

  const float pax = pos[a * 3 + 0], pay = pos[a * 3 + 1], paz = pos[a * 3 + 2];
  if (tid < ROWS) {
    const int bAtom = colBase + tid;
    float dx = pos[bAtom * 3 + 0] - pax;
    float dy = pos[bAtom * 3 + 1] - pay;
    float dz = pos[bAtom * 3 + 2] - paz;
    float r = sqrtf(dx * dx + dy * dy + dz * dz + 1e-12f);
    float iv = 1.0f / fmaxf(r, 1e-8f);
    geo[tid * 8 + 0] = dx * iv;
    geo[tid * 8 + 1] = dy * iv;
    geo[tid * 8 + 2] = dz * iv;
    geo[tid * 8 + 3] = r;
    float t0 = r * 0.666666667f;             // (r-0)/1.5
    float t1 = (r - 1.5f) * 0.666666667f;
    float t2 = (r - 3.0f) * 0.666666667f;
    geo[tid * 8 + 4] = expf(-t0 * t0);
    geo[tid * 8 + 5] = expf(-t1 * t1);
    geo[tid * 8 + 6] = expf(-t2 * t2);
  }
  __syncthreads();

  run_radial_mlp<3>(hA, hB, Wp, bsh, geo, W1, b1, W2, b2, W3, b3, W4, b4, 44,
                    tid, wave, lane);

  // epilogue: per-pair tensor contraction (R.x and R.x.Y), then block reduce over b
  float* red = hA;  // reuse: ROWS*44
  if (tid < ROWS) {
    float dx = geo[tid * 8 + 0], dy = geo[tid * 8 + 1], dz = geo[tid * 8 + 2];
    float r = geo[tid * 8 + 3];
    bool mask = r > 1e-8f;
    float Y[36];
    compute_sph(dx, dy, dz, mask, Y);
    const int bAtom = colBase + tid;
    float x0 = x[bAtom * 4 + 0], x1 = x[bAtom * 4 + 1];
    float x2 = x[bAtom * 4 + 2], x3 = x[bAtom * 4 + 3];
    const float* R = &hB[tid * HID_P];
    float out[41];
#pragma unroll
    for (int u = 0; u < 6; ++u)
      out[u] = R[u*4+0]*x0 + R[u*4+1]*x1 + R[u*4+2]*x2 + R[u*4+3]*x3;
    int o = 6;
#pragma unroll
    for (int l = 1; l <= 5; ++l) {
      const float* Rl = R + 24 + (l - 1) * 4;
      float t = Rl[0]*x0 + Rl[1]*x1 + Rl[2]*x2 + Rl[3]*x3;
#pragma unroll
      for (int m = 0; m < 2*l+1; ++m) out[o + m] = t * Y[IOFF[l] + m];
      o += 2*l+1;
    }
#pragma unroll
    for (int j = 0; j < 41; ++j) red[tid * 44 + j] = out[j];
  }
  __syncthreads();
  if (tid < 41) {
    float s = 0.0f;
    for (int p = 0; p < ROWS; ++p) s += red[p * 44 + tid];
    atomicAdd(&h_pre[a * 44 + tid], s);
  }
}

// ================= kernel 2: normalization + gated nonlinearity =================
__global__ void gate_kernel(const float* __restrict__ h_pre, float* __restrict__ h) {
  int a = blockIdx.x * blockDim.x + threadIdx.x;
  if (a >= N_ATOMS) return;
  const float inv = 0.03608439182435161f;  // 1/sqrt(768)
  h[a * 36 + 0] = ssp(h_pre[a * 44 + 0] * inv);
  int src = 6, dst = 1;
#pragma unroll
  for (int l = 1; l <= 5; ++l) {
    float g  = 1.0f / (1.0f + expf(-h_pre[a * 44 + l] * inv));
    float sc = inv * RSQL[l] * g;
#pragma unroll
    for (int m = 0; m < 2*l+1; ++m)
      h[a * 36 + dst + m] = h_pre[a * 44 + src + m] * sc;
    src += 2*l+1; dst += 2*l+1;
  }
}

// ================= kernel 3/4: fused conv_scalar_out over gathered subset =================
__global__ void __launch_bounds__(BLOCK_THREADS)
conv_out_kernel(const int* __restrict__ idx, int nSub,
                const float* __restrict__ pos, const float* __restrict__ h, // [768,36]
                const float* W1, const float* b1, const float* W2, const float* b2,
                const float* W3, const float* b3, const float* W4, const float* b4,
                float* __restrict__ out) {  // accumulated, one scalar per subset row
  extern __shared__ float sm[];
  float* hA  = sm;
  float* hB  = hA + ROWS * HID_P;
  v2f*   Wp  = (v2f*)(hB + ROWS * HID_P);
  float* bsh = (float*)(Wp + K2 * HID);
  float* geo = bsh + HID;

  const int tid = threadIdx.x, wave = tid >> 5, lane = tid & 31;
  const int aRow = blockIdx.y;
  const int colBase = blockIdx.x * ROWS;
  const int aAtom = idx[aRow];

  const float pax = pos[aAtom * 3 + 0], pay = pos[aAtom * 3 + 1], paz = pos[aAtom * 3 + 2];
  if (tid < ROWS) {
    int bcol = colBase + tid; if (bcol >= nSub) bcol = nSub - 1;
    const int bAtom = idx[bcol];
    float dx = pos[bAtom * 3 + 0] - pax;
    float dy = pos[bAtom * 3 + 1] - pay;
    float dz = pos[bAtom * 3 + 2] - paz;
    float r = sqrtf(dx * dx + dy * dy + dz * dz + 1e-12f);
    float iv = 1.0f / fmaxf(r, 1e-8f);
    geo[tid * 8 + 0] = dx * iv;
    geo[tid * 8 + 1] = dy * iv;
    geo[tid * 8 + 2] = dz * iv;
    geo[tid * 8 + 3] = r;
    float t0 = r * 0.666666667f;
    float t1 = (r - 1.5f) * 0.666666667f;
    float t2 = (r - 3.0f) * 0.666666667f;
    geo[tid * 8 + 4] = expf(-t0 * t0);
    geo[tid * 8 + 5] = expf(-t1 * t1);
    geo[tid * 8 + 6] = expf(-t2 * t2);
  }
  __syncthreads();

  run_radial_mlp<1>(hA, hB, Wp, bsh, geo, W1, b1, W2, b2, W3, b3, W4, b4, 6,
                    tid, wave, lane);

  float* red = hA;  // ROWS scalars
  if (tid < ROWS) {
    float contrib = 0.0f;
    if (colBase + tid < nSub) {
      float dx = geo[tid * 8 + 0], dy = geo[tid * 8 + 1], dz = geo[tid * 8 + 2];
      float r = geo[tid * 8 + 3];
      bool mask = r > 1e-8f;
      float Y[36];
      compute_sph(dx, dy, dz, mask, Y);
      const int bAtom = idx[colBase + tid];
      const float* f = &h[bAtom * 36];
      const float* R = &hB[tid * HID_P];
      contrib = R[0] * f[0];
#pragma unroll
      for (int l = 1; l <= 5; ++l) {
        float s = 0.0f;
#pragma unroll
        for (int m = 0; m < 2*l+1; ++m) s += f[IOFF[l] + m] * Y[IOFF[l] + m];
        contrib += R[l] * s * RSQL[l];
      }
    }
    red[tid] = contrib;
  }
  __syncthreads();
  for (int stp = ROWS / 2; stp > 0; stp >>= 1) {
    if (tid < stp) red[tid] += red[tid + stp];
    __syncthreads();
  }
  if (tid == 0) atomicAdd(&out[aRow], red[0]);
}

__global__ void zero_kernel(float* p, int n) {
  int i = blockIdx.x * blockDim.x + threadIdx.x;
  if (i < n) p[i] = 0.0f;
}

// ================= host launcher =================
extern "C" void kernel_launch(void* const* d_in, const int* in_sizes, int n_in,
                              void* d_out, int out_size, void* d_ws, size_t ws_size,
                              hipStream_t stream) {
  const float* x    = (const float*)d_in[0];   // [1,768,4]
  const float* pos  = (const float*)d_in[1];   // [1,768,3]
  const int*   idxO = (const int*)d_in[2];     // [256]
  const int*   idxH = (const int*)d_in[3];     // [512]
  // rad1_params: 4 (W,b) pairs
  const float* r1W1 = (const float*)d_in[4];  const float* r1b1 = (const float*)d_in[5];
  const float* r1W2 = (const float*)d_in[6];  const float* r1b2 = (const float*)d_in[7];
  const float* r1W3 = (const float*)d_in[8];  const float* r1b3 = (const float*)d_in[9];
  const float* r1W4 = (const float*)d_in[10]; const float* r1b4 = (const float*)d_in[11];
  const float* oW1  = (const float*)d_in[12]; const float* ob1  = (const float*)d_in[13];
  const float* oW2  = (const float*)d_in[14]; const float* ob2  = (const float*)d_in[15];
  const float* oW3  = (const float*)d_in[16]; const float* ob3  = (const float*)d_in[17];
  const float* oW4  = (const float*)d_in[18]; const float* ob4  = (const float*)d_in[19];
  const float* hW1  = (const float*)d_in[20]; const float* hb1  = (const float*)d_in[21];
  const float* hW2  = (const float*)d_in[22]; const float* hb2  = (const float*)d_in[23];
  const float* hW3  = (const float*)d_in[24]; const float* hb3  = (const float*)d_in[25];
  const float* hW4  = (const float*)d_in[26]; const float* hb4  = (const float*)d_in[27];
  (void)in_sizes; (void)n_in; (void)ws_size;

  float* h_pre = (float*)d_ws;                 // [768,44]
  float* hfeat = h_pre + N_ATOMS * 44;         // [768,36]
  float* out   = (float*)d_out;                // [256]+[512]

  const size_t smem = (size_t)SMEM_FLOATS * sizeof(float);
  hipFuncSetAttribute((const void*)rad1_conv_in_kernel,
                      hipFuncAttributeMaxDynamicSharedMemorySize, (int)smem);
  hipFuncSetAttribute((const void*)conv_out_kernel,
                      hipFuncAttributeMaxDynamicSharedMemorySize, (int)smem);

  zero_kernel<<<(N_ATOMS * 44 + 255) / 256, 256, 0, stream>>>(h_pre, N_ATOMS * 44);
  zero_kernel<<<(out_size + 255) / 256, 256, 0, stream>>>(out, out_size);

  rad1_conv_in_kernel<<<dim3(N_ATOMS / ROWS, N_ATOMS), BLOCK_THREADS, smem, stream>>>(
      x, pos, r1W1, r1b1, r1W2, r1b2, r1W3, r1b3, r1W4, r1b4, h_pre);

  gate_kernel<<<(N_ATOMS + 255) / 256, 256, 0, stream>>>(h_pre, hfeat);

  conv_out_kernel<<<dim3(256 / ROWS, 256), BLOCK_THREADS, smem, stream>>>(
      idxO, 256, pos, hfeat, oW1, ob1, oW2, ob2, oW3, ob3, oW4, ob4, out);

  conv_out_kernel<<<dim3(512 / ROWS, 512), BLOCK_THREADS, smem, stream>>>(
      idxH, 512, pos, hfeat, hW1, hb1, hW2, hb2, hW3, hb3, hW4, hb4, out + 256);
}